// VolumeRenderer_30185030156483
// MI455X (gfx1250) — compile-verified
//
#include <hip/hip_runtime.h>

// Volume rendering (NeRF-style ray marching), R=131072 rays, S=128 samples, C=3.
// Memory-bound: ~337 MB single-pass traffic -> ~14.5 us at 23.3 TB/s on MI455X.
// No matmul structure -> WMMA inapplicable; optimize the data-movement path:
//  - one wave32 per ray, one global_load_b128 per lane per row (perfect coalescing)
//  - non-temporal loads everywhere (single-pass stream > 192MB L2)
//  - gfx1250 async global->LDS copy of the feature row (ASYNCcnt path),
//    issued before the transmittance scan and waited with s_wait_asynccnt
//  - 5-step wave32 shuffle scan (inclusive cumsum) + butterfly reductions.

#define EPS_F        1e-10f
#define FAR_DELTA_F  1e10f

typedef float v4f __attribute__((ext_vector_type(4)));
typedef int   v4i __attribute__((ext_vector_type(4)));

constexpr int S   = 128;  // samples per ray (fixed by reference)
constexpr int C   = 3;    // channels (fixed by reference)
constexpr int WPB = 8;    // waves (rays) per block
constexpr int BLOCK = WPB * 32;

#if defined(__gfx1250__) && __has_builtin(__builtin_amdgcn_global_load_async_to_lds_b128)
#define USE_ASYNC_LDS 1
// Exact parameter types per clang's prototype "vV4i*1V4i*3IiIi":
typedef __attribute__((address_space(1))) v4i* gbl_v4i_p;  // global src
typedef __attribute__((address_space(3))) v4i* lds_v4i_p;  // LDS dst
// gfx12 CPol: TH in bits [2:0]; TH_NT = 1 (non-temporal, keep L2 clean for the stream)
#define ASYNC_CPOL 1
#else
#define USE_ASYNC_LDS 0
#endif

__global__ __launch_bounds__(BLOCK)
void volrend_kernel(const float* __restrict__ density,
                    const float* __restrict__ feature,
                    const float* __restrict__ depth,
                    float* __restrict__ feat_out,   // [R,3]
                    float* __restrict__ depth_out,  // [R]
                    int R)
{
#if USE_ASYNC_LDS
    __shared__ __align__(16) float lfeat[WPB][S * C];  // 12 KB: feature staging
#endif
    const int lane = threadIdx.x & 31;
    const int wave = threadIdx.x >> 5;
    const int ray  = blockIdx.x * WPB + wave;
    if (ray >= R) return;  // uniform per wave; no barriers used below

    const size_t rbase = (size_t)ray * S;

#if USE_ASYNC_LDS
    // Kick off the feature-row fetch (1536 B) into LDS before the scan math.
    {
        const float* fsrc = feature + rbase * C;
        #pragma unroll
        for (int i = 0; i < 3; ++i) {
            const int off = i * 128 + lane * 4;  // floats
            __builtin_amdgcn_global_load_async_to_lds_b128(
                (gbl_v4i_p)(const void*)(fsrc + off),
                (lds_v4i_p)(void*)(&lfeat[wave][off]),
                0, ASYNC_CPOL);
        }
    }
#endif

    // Lane l owns samples 4l..4l+3. Single b128 load per row per lane.
    const v4f d4 = __builtin_nontemporal_load((const v4f*)(depth   + rbase) + lane);
    const v4f s4 = __builtin_nontemporal_load((const v4f*)(density + rbase) + lane);

    // deltas: need first depth of the next lane's chunk for the 4th delta.
    const float nextd = __shfl_down(d4.x, 1, 32);          // lane 31: unused
    const float del0 = d4.y - d4.x;
    const float del1 = d4.z - d4.y;
    const float del2 = d4.w - d4.z;
    const float del3 = (lane == 31) ? FAR_DELTA_F : (nextd - d4.w);

    const float sd0 = s4.x * del0;
    const float sd1 = s4.y * del1;
    const float sd2 = s4.z * del2;
    const float sd3 = s4.w * del3;

    // 4-wide serial prefix inside the lane...
    const float p0 = sd0;
    const float p1 = p0 + sd1;
    const float p2 = p1 + sd2;
    const float p3 = p2 + sd3;

    // ...then 5-step wave32 inclusive scan of the lane totals.
    float sum = p3;
    #pragma unroll
    for (int off = 1; off < 32; off <<= 1) {
        const float t = __shfl_up(sum, off, 32);
        if (lane >= off) sum += t;
    }
    const float excl = sum - p3;  // exclusive prefix for this lane's chunk

    // weights: T_i = exp(-inclusive cumsum), alpha_i = 1 - exp(-sd_i + eps)
    const float w0 = __expf(-(excl + p0)) * (1.0f - __expf(-sd0 + EPS_F));
    const float w1 = __expf(-(excl + p1)) * (1.0f - __expf(-sd1 + EPS_F));
    const float w2 = __expf(-(excl + p2)) * (1.0f - __expf(-sd2 + EPS_F));
    const float w3 = __expf(-(excl + p3)) * (1.0f - __expf(-sd3 + EPS_F));

    // Feature data: 12 floats per lane = 3 x b128, all 16B-aligned.
#if USE_ASYNC_LDS
#if __has_builtin(__builtin_amdgcn_s_wait_asynccnt)
    __builtin_amdgcn_s_wait_asynccnt(0);
#else
    asm volatile("s_wait_asynccnt 0" ::: "memory");
#endif
    asm volatile("" ::: "memory");  // keep LDS reads below the wait
    const v4f* lf4 = (const v4f*)(&lfeat[wave][lane * 12]);
    const v4f fa = lf4[0];
    const v4f fb = lf4[1];
    const v4f fc = lf4[2];
#else
    const v4f* lf4 = (const v4f*)(feature + rbase * C + (size_t)lane * 12);
    const v4f fa = __builtin_nontemporal_load(lf4 + 0);
    const v4f fb = __builtin_nontemporal_load(lf4 + 1);
    const v4f fc = __builtin_nontemporal_load(lf4 + 2);
#endif

    // samples (ch0,ch1,ch2): s0=(fa.x,fa.y,fa.z) s1=(fa.w,fb.x,fb.y)
    //                        s2=(fb.z,fb.w,fc.x) s3=(fc.y,fc.z,fc.w)
    float f0 = w0 * fa.x + w1 * fa.w + w2 * fb.z + w3 * fc.y;
    float f1 = w0 * fa.y + w1 * fb.x + w2 * fb.w + w3 * fc.z;
    float f2 = w0 * fa.z + w1 * fb.y + w2 * fc.x + w3 * fc.w;
    float dd = w0 * d4.x + w1 * d4.y + w2 * d4.z + w3 * d4.w;

    // Butterfly reduction across the wave (all lanes end with the total).
    #pragma unroll
    for (int off = 16; off > 0; off >>= 1) {
        f0 += __shfl_xor(f0, off, 32);
        f1 += __shfl_xor(f1, off, 32);
        f2 += __shfl_xor(f2, off, 32);
        dd += __shfl_xor(dd, off, 32);
    }

    if (lane == 0) {
        feat_out[(size_t)ray * 3 + 0] = f0;
        feat_out[(size_t)ray * 3 + 1] = f1;
        feat_out[(size_t)ray * 3 + 2] = f2;
        depth_out[ray]                = dd;
    }
}

extern "C" void kernel_launch(void* const* d_in, const int* in_sizes, int n_in,
                              void* d_out, int out_size, void* d_ws, size_t ws_size,
                              hipStream_t stream) {
    const float* density = (const float*)d_in[0];  // [R,S]
    const float* feature = (const float*)d_in[1];  // [R,S,C]
    const float* depth   = (const float*)d_in[2];  // [R,S]
    const int R = in_sizes[0] / S;

    float* feat_out  = (float*)d_out;              // first R*3 floats
    float* depth_out = feat_out + (size_t)R * C;   // next R floats

    const int grid = (R + WPB - 1) / WPB;
    hipLaunchKernelGGL(volrend_kernel, dim3(grid), dim3(BLOCK), 0, stream,
                       density, feature, depth, feat_out, depth_out, R);
}